// GAT_65068754534588
// MI455X (gfx1250) — compile-verified
//
#include <hip/hip_runtime.h>
#include <math.h>

// ---- problem constants (match reference) ----
#define FIN 128
#define HC 256      // H*C, layer-1 hidden
#define NH 8        // heads layer 1
#define NC 32       // channels per head layer 1
#define NCLS 16     // classes (layer-2 width, H=1)
#define NG 64       // graphs
#define NEG_SLOPE 0.2f

#define LDS_STRIDE 132   // 16 A-rows x 132 floats: bank = (4r+k)%64 -> conflict-free, 8B aligned

typedef __attribute__((ext_vector_type(2))) float v2f;
typedef __attribute__((ext_vector_type(8))) float v8f;

__device__ __forceinline__ float lrelu(float x) { return x > 0.f ? x : NEG_SLOPE * x; }

// order-preserving float <-> uint key for atomicMax-based segment max
__device__ __forceinline__ unsigned f2key(float f) {
    unsigned u = __float_as_uint(f);
    return (u & 0x80000000u) ? ~u : (u | 0x80000000u);
}
__device__ __forceinline__ float key2f(unsigned k) {
    unsigned u = (k & 0x80000000u) ? (k & 0x7fffffffu) : ~k;
    return __uint_as_float(u);
}

// ---------------------------------------------------------------------------
// GEMM1: D[M,256] = A[M,128] @ B[128,256], fp32 WMMA, LDS-staged A.
// One 256-thread workgroup (8 waves) per 16-row M strip. The 16x128 A tile is
// loaded to LDS once (coalesced), then each wave computes two adjacent 16x16
// output tiles (tn = 2*wave, 2*wave+1), sharing the A registers across two
// V_WMMA_F32_16X16X4_F32 issues per K-step.
// A 16x4 per-lane layout: lanes 0-15 row M=lane, VGPR0/1 = K{0,1}; lanes 16-31 K{2,3}.
// B 4x16: lanes 0-15 col N=lane, VGPR0/1 = K{0,1}; lanes 16-31 K{2,3}.
// D: VGPR i -> (M = (lane>=16?8:0)+i, N = lane&15).
// ---------------------------------------------------------------------------
__global__ void wmma_gemm1_lds(const float* __restrict__ A, const float* __restrict__ B,
                               float* __restrict__ D, int M) {
    __shared__ float As[16 * LDS_STRIDE];

    const int m0 = blockIdx.x * 16;

    // stage A tile (16 x 128) into LDS, coalesced, rows clamped for ragged M
    for (int idx = threadIdx.x; idx < 16 * FIN; idx += 256) {
        int row = idx >> 7;          // / FIN
        int col = idx & (FIN - 1);
        int gr = m0 + row; if (gr >= M) gr = M - 1;
        As[row * LDS_STRIDE + col] = A[(size_t)gr * FIN + col];
    }
    __syncthreads();

    const int lane = threadIdx.x & 31;
    const int wave = threadIdx.x >> 5;           // 0..7
    const int tn0  = wave * 2;                   // this wave's two N tiles
    const int r    = lane & 15;
    const int kh   = (lane >> 4) << 1;           // 0 or 2

    const int lbase = r * LDS_STRIDE + kh;
    const float* bp0 = B + (size_t)kh * HC + tn0 * 16 + r;
    const float* bp1 = bp0 + 16;

    v8f acc0 = {}, acc1 = {};
    for (int kk = 0; kk < FIN; kk += 4) {
        v2f a, b0, b1;
        a.x  = As[lbase + kk];
        a.y  = As[lbase + kk + 1];
        b0.x = bp0[0]; b0.y = bp0[HC];
        b1.x = bp1[0]; b1.y = bp1[HC];
        bp0 += 4 * HC;
        bp1 += 4 * HC;
        acc0 = __builtin_amdgcn_wmma_f32_16x16x4_f32(false, a, false, b0,
                                                     (short)0, acc0, false, false);
        acc1 = __builtin_amdgcn_wmma_f32_16x16x4_f32(false, a, false, b1,
                                                     (short)0, acc1, false, false);
    }

    const int rb = m0 + ((lane >> 4) << 3);
    const int c0 = tn0 * 16 + (lane & 15);
#pragma unroll
    for (int i = 0; i < 8; ++i) {
        int row = rb + i;
        if (row < M) {
            D[(size_t)row * HC + c0]      = acc0[i];
            D[(size_t)row * HC + c0 + 16] = acc1[i];
        }
    }
}

// ---------------------------------------------------------------------------
// Generic fp32 WMMA GEMM (used for GEMM2: [M,256]@[256,16], single N tile).
// One wave32 per 16x16 output tile, direct global loads.
// ---------------------------------------------------------------------------
__global__ void wmma_gemm_f32(const float* __restrict__ A, const float* __restrict__ B,
                              float* __restrict__ D, int M, int K, int N) {
    const int lane = threadIdx.x & 31;
    const int wave = blockIdx.x * (blockDim.x >> 5) + (threadIdx.x >> 5);
    const int ntn  = N >> 4;
    const int tm   = wave / ntn;
    const int tn   = wave - tm * ntn;
    if (tm * 16 >= M) return;                  // uniform across the wave

    const int r  = lane & 15;
    const int kh = (lane >> 4) << 1;           // 0 or 2
    int ar = tm * 16 + r; if (ar >= M) ar = M - 1;

    const float* ap = A + (size_t)ar * K + kh;
    const float* bp = B + (size_t)kh * N + tn * 16 + r;

    v8f acc = {};
    for (int kk = 0; kk < K; kk += 4) {
        v2f a, b;
        a.x = ap[0]; a.y = ap[1];
        b.x = bp[0]; b.y = bp[N];
        ap += 4;
        bp += (size_t)4 * N;
        acc = __builtin_amdgcn_wmma_f32_16x16x4_f32(false, a, false, b,
                                                    (short)0, acc, false, false);
    }

    const int col = tn * 16 + (lane & 15);
    const int rb  = tm * 16 + ((lane >> 4) << 3);
#pragma unroll
    for (int i = 0; i < 8; ++i) {
        int row = rb + i;
        if (row < M) D[(size_t)row * N + col] = acc[i];
    }
}

// al_s[n,h] = <xp[n,h,:], a_src[h,:]>, al_d likewise. One thread per (node,head).
__global__ void attn_logits1(const float* __restrict__ xp, const float* __restrict__ a_src,
                             const float* __restrict__ a_dst, float* __restrict__ als,
                             float* __restrict__ ald, int n) {
    int i = blockIdx.x * blockDim.x + threadIdx.x;
    if (i >= n * NH) return;
    int h = i % NH;
    const float* row = xp + (size_t)(i / NH) * HC + h * NC;
    float s = 0.f, d = 0.f;
#pragma unroll
    for (int c = 0; c < NC; ++c) {
        float v = row[c];
        s += v * a_src[h * NC + c];
        d += v * a_dst[h * NC + c];
    }
    als[i] = s; ald[i] = d;
}

// One thread per node (layer 2, H=1, C=NCLS).
__global__ void attn_logits2(const float* __restrict__ xp2, const float* __restrict__ a_src,
                             const float* __restrict__ a_dst, float* __restrict__ als,
                             float* __restrict__ ald, int n) {
    int i = blockIdx.x * blockDim.x + threadIdx.x;
    if (i >= n) return;
    const float* row = xp2 + (size_t)i * NCLS;
    float s = 0.f, d = 0.f;
#pragma unroll
    for (int c = 0; c < NCLS; ++c) {
        float v = row[c];
        s += v * a_src[c];
        d += v * a_dst[c];
    }
    als[i] = s; ald[i] = d;
}

__device__ __forceinline__ void edge_endpoints(const int* ei, int E, int e, int& src, int& dst) {
    if (e < E) { src = ei[e]; dst = ei[E + e]; }
    else       { src = dst = e - E; }          // appended self-loops
}

// segment max over dst: one thread per (edge, head)
__global__ void edge_max_k(const int* __restrict__ ei, int E, int n, int nh,
                           const float* __restrict__ als, const float* __restrict__ ald,
                           unsigned* __restrict__ mkey) {
    long i = (long)blockIdx.x * blockDim.x + threadIdx.x;
    long tot = (long)(E + n) * nh;
    if (i >= tot) return;
    int e = (int)(i / nh), h = (int)(i - (long)e * nh);
    int src, dst; edge_endpoints(ei, E, e, src, dst);
    float v = lrelu(als[(size_t)src * nh + h] + ald[(size_t)dst * nh + h]);
    atomicMax(&mkey[(size_t)dst * nh + h], f2key(v));
}

// softmax denominator: one thread per (edge, head); logits recomputed (VALU-cheap)
__global__ void edge_denom_k(const int* __restrict__ ei, int E, int n, int nh,
                             const float* __restrict__ als, const float* __restrict__ ald,
                             const unsigned* __restrict__ mkey, float* __restrict__ denom) {
    long i = (long)blockIdx.x * blockDim.x + threadIdx.x;
    long tot = (long)(E + n) * nh;
    if (i >= tot) return;
    int e = (int)(i / nh), h = (int)(i - (long)e * nh);
    int src, dst; edge_endpoints(ei, E, e, src, dst);
    float v = lrelu(als[(size_t)src * nh + h] + ald[(size_t)dst * nh + h]);
    atomicAdd(&denom[(size_t)dst * nh + h], expf(v - key2f(mkey[(size_t)dst * nh + h])));
}

// layer-1 aggregation: one 256-thread block per edge; thread t = (head t/32, chan t%32)
// coalesced gather of xp1[src,:] and coalesced 256-float atomic burst into agg[dst,:]
__global__ void edge_scatter1(const int* __restrict__ ei, int E, int n,
                              const float* __restrict__ als, const float* __restrict__ ald,
                              const unsigned* __restrict__ mkey, const float* __restrict__ denom,
                              const float* __restrict__ xp, float* __restrict__ agg) {
    int e = blockIdx.x;
    int t = threadIdx.x;
    int h = t >> 5;
    int src, dst; edge_endpoints(ei, E, e, src, dst);
    float v = lrelu(als[(size_t)src * NH + h] + ald[(size_t)dst * NH + h]);
    float alpha = expf(v - key2f(mkey[(size_t)dst * NH + h])) / denom[(size_t)dst * NH + h];
    atomicAdd(&agg[(size_t)dst * HC + t], xp[(size_t)src * HC + t] * alpha);
}

// layer-2 aggregation: one thread per (edge, class)
__global__ void edge_scatter2(const int* __restrict__ ei, int E, int n,
                              const float* __restrict__ als, const float* __restrict__ ald,
                              const unsigned* __restrict__ mkey, const float* __restrict__ denom,
                              const float* __restrict__ xp2, float* __restrict__ agg2) {
    long i = (long)blockIdx.x * blockDim.x + threadIdx.x;
    long tot = (long)(E + n) * NCLS;
    if (i >= tot) return;
    int e = (int)(i >> 4), c = (int)(i & 15);
    int src, dst; edge_endpoints(ei, E, e, src, dst);
    float v = lrelu(als[src] + ald[dst]);
    float alpha = expf(v - key2f(mkey[dst])) / denom[dst];
    atomicAdd(&agg2[(size_t)dst * NCLS + c], xp2[(size_t)src * NCLS + c] * alpha);
}

// h1 = elu(agg1 + b1), written back into the (now free) xp1 buffer
__global__ void elu_bias(const float* __restrict__ agg, const float* __restrict__ b1,
                         float* __restrict__ h1, int n) {
    long i = (long)blockIdx.x * blockDim.x + threadIdx.x;
    if (i >= (long)n * HC) return;
    float v = agg[i] + b1[i & (HC - 1)];
    h1[i] = v > 0.f ? v : (expf(v) - 1.f);
}

// mean-pool inputs: pooled[g,c] += h2[n,c]; counts[g] += 1
__global__ void pool_k(const float* __restrict__ agg2, const float* __restrict__ b2,
                       const int* __restrict__ batch, float* __restrict__ pooled,
                       float* __restrict__ counts, int n) {
    long i = (long)blockIdx.x * blockDim.x + threadIdx.x;
    if (i >= (long)n * NCLS) return;
    int node = (int)(i >> 4), c = (int)(i & 15);
    int g = batch[node];
    atomicAdd(&pooled[(size_t)g * NCLS + c], agg2[i] + b2[c]);
    if (c == 0) atomicAdd(&counts[g], 1.0f);
}

// divide by count, log_softmax per graph; one thread per graph (G=64)
__global__ void finalize_k(const float* __restrict__ pooled, const float* __restrict__ counts,
                           float* __restrict__ out) {
    int g = threadIdx.x;
    if (g >= NG) return;
    float cnt = counts[g]; cnt = cnt > 1.f ? cnt : 1.f;
    float v[NCLS];
    float m = -INFINITY;
#pragma unroll
    for (int c = 0; c < NCLS; ++c) { v[c] = pooled[g * NCLS + c] / cnt; m = fmaxf(m, v[c]); }
    float s = 0.f;
#pragma unroll
    for (int c = 0; c < NCLS; ++c) s += expf(v[c] - m);
    float lse = logf(s);
#pragma unroll
    for (int c = 0; c < NCLS; ++c) out[g * NCLS + c] = v[c] - m - lse;
}

static inline unsigned cdiv(long a, long b) { return (unsigned)((a + b - 1) / b); }

extern "C" void kernel_launch(void* const* d_in, const int* in_sizes, int n_in,
                              void* d_out, int out_size, void* d_ws, size_t ws_size,
                              hipStream_t stream) {
    const float* x    = (const float*)d_in[0];
    const int*   ei   = (const int*)d_in[1];
    const int*   bat  = (const int*)d_in[2];
    const float* W1   = (const float*)d_in[3];
    const float* a1s  = (const float*)d_in[4];
    const float* a1d  = (const float*)d_in[5];
    const float* b1   = (const float*)d_in[6];
    const float* W2   = (const float*)d_in[7];
    const float* a2s  = (const float*)d_in[8];
    const float* a2d  = (const float*)d_in[9];
    const float* b2   = (const float*)d_in[10];
    float* out = (float*)d_out;

    const int n  = in_sizes[2];       // 50000 nodes (batch length)
    const int E  = in_sizes[1] / 2;   // 800000 edges
    const int ET = E + n;             // + self loops

    // ---- workspace carve-out (256B aligned); zeroed region is contiguous ----
    char* ws = (char*)d_ws;
    size_t off = 0;
    auto carve = [&](size_t bytes) -> void* {
        void* p = ws + off;
        off = (off + bytes + 255) & ~(size_t)255;
        return p;
    };
    float*    xp1    = (float*)carve((size_t)n * HC * 4);    // layer-1 features; later h1
    float*    xp2    = (float*)carve((size_t)n * NCLS * 4);
    float*    al1s   = (float*)carve((size_t)n * NH * 4);
    float*    al1d   = (float*)carve((size_t)n * NH * 4);
    float*    al2s   = (float*)carve((size_t)n * 4);
    float*    al2d   = (float*)carve((size_t)n * 4);
    size_t zbeg = off;
    float*    agg1   = (float*)carve((size_t)n * HC * 4);
    unsigned* m1     = (unsigned*)carve((size_t)n * NH * 4);
    float*    den1   = (float*)carve((size_t)n * NH * 4);
    float*    agg2   = (float*)carve((size_t)n * NCLS * 4);
    unsigned* m2     = (unsigned*)carve((size_t)n * 4);
    float*    den2   = (float*)carve((size_t)n * 4);
    float*    pooled = (float*)carve((size_t)NG * NCLS * 4);
    float*    counts = (float*)carve((size_t)NG * 4);
    size_t zend = off;

    hipMemsetAsync(ws + zbeg, 0, zend - zbeg, stream);  // accumulators + max keys (-inf)

    // ---- layer 1 ----
    // xp1 = x @ W1   [n,128]@[128,256]; LDS-staged A, 8 waves x 2 N-tiles per WG
    wmma_gemm1_lds<<<cdiv(n, 16), 256, 0, stream>>>(x, W1, xp1, n);
    attn_logits1<<<cdiv((long)n * NH, 256), 256, 0, stream>>>(xp1, a1s, a1d, al1s, al1d, n);
    edge_max_k  <<<cdiv((long)ET * NH, 256), 256, 0, stream>>>(ei, E, n, NH, al1s, al1d, m1);
    edge_denom_k<<<cdiv((long)ET * NH, 256), 256, 0, stream>>>(ei, E, n, NH, al1s, al1d, m1, den1);
    edge_scatter1<<<(unsigned)ET, 256, 0, stream>>>(ei, E, n, al1s, al1d, m1, den1, xp1, agg1);
    elu_bias<<<cdiv((long)n * HC, 256), 256, 0, stream>>>(agg1, b1, xp1, n);  // xp1 := h1

    // ---- layer 2 ----
    {   // xp2 = h1 @ W2   [n,256]@[256,16]; single N tile -> plain WMMA kernel
        long tiles = (long)cdiv(n, 16) * (NCLS / 16);
        wmma_gemm_f32<<<cdiv(tiles, 8), 256, 0, stream>>>(xp1, W2, xp2, n, HC, NCLS);
    }
    attn_logits2<<<cdiv(n, 256), 256, 0, stream>>>(xp2, a2s, a2d, al2s, al2d, n);
    edge_max_k  <<<cdiv((long)ET, 256), 256, 0, stream>>>(ei, E, n, 1, al2s, al2d, m2);
    edge_denom_k<<<cdiv((long)ET, 256), 256, 0, stream>>>(ei, E, n, 1, al2s, al2d, m2, den2);
    edge_scatter2<<<cdiv((long)ET * NCLS, 256), 256, 0, stream>>>(ei, E, n, al2s, al2d, m2, den2, xp2, agg2);

    // ---- pooling + log_softmax ----
    pool_k<<<cdiv((long)n * NCLS, 256), 256, 0, stream>>>(agg2, b2, bat, pooled, counts, n);
    finalize_k<<<1, 64, 0, stream>>>(pooled, counts, out);
}